// PEEModule_18442589570207
// MI455X (gfx1250) — compile-verified
//
#include <hip/hip_runtime.h>

typedef float v2f __attribute__((ext_vector_type(2)));
typedef float v8f __attribute__((ext_vector_type(8)));

#define B_  16
#define C_  512
#define O_  128
#define HW  96
#define P_  9216   // 96*96
#define KC  64     // K chunk staged in LDS
#define PAN 1040   // float offset of second 16-pixel LDS panel (== 16 mod 64)

// ---------------------------------------------------------------------------
// Kernel 1: feat[b,o,p] = sum_c W[o,c] * x[b,c,p] + bias[o]
// Block = 256 threads = 8 waves; block handles a 32-pixel tile (two 16-wide
// N-tiles -> two independent accumulator chains per wave, 2x A reuse).
// Wave w handles output rows o0 = 16w .. 16w+15.  K staged via LDS in 64-chunks.
// ---------------------------------------------------------------------------
__global__ __launch_bounds__(256)
void gemm_feat_kernel(const float* __restrict__ x, const float* __restrict__ Wm,
                      const float* __restrict__ bias, float* __restrict__ feat) {
    // panel0: [c*16 + n], panel1: [PAN + c*16 + n]; gap keeps banks conflict-free
    __shared__ float lx[PAN + KC * 16];

    const int tid  = threadIdx.x;
    const int wave = tid >> 5;
    const int lane = tid & 31;
    const int n    = lane & 15;          // A: M row index, B: N col index
    const int kh   = (lane >> 4) << 1;   // 0 or 2 (K half select per ISA layout)

    const int tile  = blockIdx.x;            // 0..4607
    const int batch = tile / (P_ / 32);
    const int pbase = (tile % (P_ / 32)) * 32;
    const float* xb = x + (size_t)batch * C_ * P_ + pbase;
    const int o0 = wave << 4;
    const float* wrow = Wm + (size_t)(o0 + n) * C_;

    v8f acc0 = {}, acc1 = {};
    for (int k0 = 0; k0 < C_; k0 += KC) {
        // cooperatively stage x[k0..k0+KC-1][pbase..pbase+31] into LDS
        #pragma unroll
        for (int i = 0; i < (KC * 32) / 256; ++i) {
            int e  = tid + i * 256;
            int c  = e >> 5, nn = e & 31;
            lx[c * 16 + (nn & 15) + ((nn >> 4) * PAN)] =
                xb[(size_t)(k0 + c) * P_ + nn];
        }
        __syncthreads();

        #pragma unroll
        for (int ks = 0; ks < KC / 4; ++ks) {
            const int k = k0 + ks * 4;
            // A-matrix 16x4 f32: lane&15 = M, VGPR0/1 = K = kh, kh+1
            float2 a2 = *reinterpret_cast<const float2*>(wrow + k + kh);
            v2f a; a.x = a2.x; a.y = a2.y;
            // B-matrix 4x16 f32: lane&15 = N, VGPR0/1 = K = kh, kh+1
            const int ci = ks * 4 + kh;
            v2f b0; b0.x = lx[ci * 16 + n];        b0.y = lx[(ci + 1) * 16 + n];
            v2f b1; b1.x = lx[PAN + ci * 16 + n];  b1.y = lx[PAN + (ci + 1) * 16 + n];
            acc0 = __builtin_amdgcn_wmma_f32_16x16x4_f32(
                false, a, false, b0, (short)0, acc0, false, false);
            acc1 = __builtin_amdgcn_wmma_f32_16x16x4_f32(
                false, a, false, b1, (short)0, acc1, false, false);
        }
        __syncthreads();
    }

    // C/D layout: VGPR r -> M = r (lanes 0-15) or 8+r (lanes 16-31), N = lane&15
    const int mbase = o0 + ((lane >> 4) << 3);
    #pragma unroll
    for (int r = 0; r < 8; ++r) {
        const float bv  = bias[mbase + r];
        const size_t row = ((size_t)batch * O_ + mbase + r) * P_ + pbase;
        feat[row + n]      = acc0[r] + bv;
        feat[row + 16 + n] = acc1[r] + bv;
    }
}

// ---------------------------------------------------------------------------
// Kernel 2: pooled means for s=1,3,6,8 per (b,o) plane.
// All cell boundaries are multiples of 4 -> each thread sums a 4x4 pixel tile
// and contributes it to exactly one cell per scale via LDS float atomics.
// Pool layout per plane (110 floats): [0]=s1, [1..9]=s3, [10..45]=s6, [46..109]=s8
// ---------------------------------------------------------------------------
__global__ __launch_bounds__(256)
void pool_kernel(const float* __restrict__ feat, float* __restrict__ pool) {
    __shared__ float pl[110];
    const int tid = threadIdx.x;
    if (tid < 110) pl[tid] = 0.0f;
    __syncthreads();

    const size_t plane = (size_t)blockIdx.x * P_;
    for (int t = tid; t < 576; t += 256) {           // 24x24 tiles of 4x4
        const int h0 = (t / 24) * 4, w0 = (t % 24) * 4;
        float s = 0.0f;
        #pragma unroll
        for (int i = 0; i < 4; ++i) {
            float4 v = *reinterpret_cast<const float4*>(feat + plane + (size_t)(h0 + i) * HW + w0);
            s += v.x + v.y + v.z + v.w;
        }
        atomicAdd(&pl[0], s);
        atomicAdd(&pl[1  + (h0 >> 5) * 3 + (w0 >> 5)], s);
        atomicAdd(&pl[10 + (h0 >> 4) * 6 + (w0 >> 4)], s);
        atomicAdd(&pl[46 + (h0 / 12) * 8 + (w0 / 12)], s);
    }
    __syncthreads();

    if (tid < 110) {
        float inv = (tid == 0)  ? (1.0f / 9216.0f)
                  : (tid < 10)  ? (1.0f / 1024.0f)
                  : (tid < 46)  ? (1.0f / 256.0f)
                  :               (1.0f / 144.0f);
        pool[(size_t)blockIdx.x * 110 + tid] = pl[tid] * inv;
    }
}

// ---------------------------------------------------------------------------
// Kernel 3: out[b, br*128+o, p] = feat[b,o,p] - bilinear_up(pool_s)[p]
// One block per (b,o) plane; pool values staged in LDS.
// ---------------------------------------------------------------------------
__global__ __launch_bounds__(256)
void out_kernel(const float* __restrict__ feat, const float* __restrict__ pool,
                float* __restrict__ out) {
    __shared__ float pl[110];
    const int tid = threadIdx.x;
    const int bo  = blockIdx.x;              // b*128 + o
    if (tid < 110) pl[tid] = pool[(size_t)bo * 110 + tid];
    __syncthreads();

    const int b = bo >> 7, o = bo & 127;
    const size_t fplane = (size_t)bo * P_;
    const size_t obase  = ((size_t)b * 512 + o) * P_;   // + br*128*P_ per branch

    const int S[4]   = {1, 3, 6, 8};
    const int OFF[4] = {0, 1, 10, 46};

    for (int p = tid; p < P_; p += 256) {
        const float f = feat[fplane + p];
        const int h = p / HW, w = p % HW;
        #pragma unroll
        for (int si = 0; si < 4; ++si) {
            const int s = S[si];
            float up;
            if (s == 1) {
                up = pl[0];
            } else {
                const float ry = (float)h * (float)(s - 1) * (1.0f / 95.0f);
                const float rx = (float)w * (float)(s - 1) * (1.0f / 95.0f);
                const int y0 = (int)ry, x0 = (int)rx;
                const int y1 = min(y0 + 1, s - 1), x1 = min(x0 + 1, s - 1);
                const float wy = ry - (float)y0, wx = rx - (float)x0;
                const float* pp = &pl[OFF[si]];
                const float tl = pp[y0 * s + x0], tr = pp[y0 * s + x1];
                const float bl = pp[y1 * s + x0], br = pp[y1 * s + x1];
                const float l = tl + (bl - tl) * wy;     // y-lerp first (as ref)
                const float r = tr + (br - tr) * wy;
                up = l + (r - l) * wx;                   // then x-lerp
            }
            out[obase + (size_t)si * O_ * P_ + p] = f - up;
        }
    }
}

extern "C" void kernel_launch(void* const* d_in, const int* in_sizes, int n_in,
                              void* d_out, int out_size, void* d_ws, size_t ws_size,
                              hipStream_t stream) {
    const float* x    = (const float*)d_in[0];   // (16,512,96,96)
    const float* Wm   = (const float*)d_in[1];   // (128,512)
    const float* bias = (const float*)d_in[2];   // (128,)
    float* out  = (float*)d_out;                 // (16,512,96,96)
    float* feat = (float*)d_ws;                  // 16*128*9216 floats
    float* pool = feat + (size_t)B_ * O_ * P_;   // 2048*110 floats

    gemm_feat_kernel<<<4608, 256, 0, stream>>>(x, Wm, bias, feat);
    pool_kernel<<<B_ * O_, 256, 0, stream>>>(feat, pool);
    out_kernel<<<B_ * O_, 256, 0, stream>>>(feat, pool, out);
}